// DownNet_mat_2ring_small_confounded_75505525063935
// MI455X (gfx1250) — compile-verified
//
#include <hip/hip_runtime.h>

typedef __attribute__((ext_vector_type(2))) float v2f;
typedef __attribute__((ext_vector_type(4))) float v4f;
typedef __attribute__((ext_vector_type(8))) float v8f;

#define K_NEIGH 19
#define LRELU_SLOPE 0.2f
#define BN_EPS 1e-5f

// ---------------------------------------------------------------------------
// Conv layer on (V, B=8, C) activations:
//   out[v,b,o] = sum_{k,c} x[neigh[v*19+k], b, c] * W[o,k,c] + bias[o]
// One wave per (6-vertex, 32-output-channel) tile:
//   2 M-tiles (A = W rows) x 3 N-tiles (B = gathered feat, 2 verts x 8 batch)
//   -> 6 WMMA f32 16x16x4 per contraction chunk fed by 2 A-loads + 3 B-loads.
// Fused per-channel sum/sumsq for the following batch-norm.
// ---------------------------------------------------------------------------
__global__ __launch_bounds__(32)
void conv_wmma_kernel(const float* __restrict__ xin,   // (V, 8, Cin)
                      const int*   __restrict__ neigh,
                      const float* __restrict__ Wt,    // (Cout, 19, Cin)
                      const float* __restrict__ bias,  // (Cout)
                      float*       __restrict__ out,   // (V, 8, Cout)
                      float*       __restrict__ sums,  // (2*Cout): [sum | sumsq]
                      int V, int Cin, int Cout)
{
    const int lane  = threadIdx.x;       // 0..31 (wave32)
    const int n     = lane & 15;         // output column 0..15
    const int half  = lane >> 4;         // contraction sub-pair select
    const int bb    = n & 7;             // batch
    const int vsel  = n >> 3;            // vertex within pair
    const int vbase = blockIdx.x * 6;
    const int KC    = K_NEIGH * Cin;
    const int obase = blockIdx.y * 32;   // first of this wave's 32 channels

    // A-matrix row pointers for the two M-tiles (per-lane row = n)
    const float* w0 = Wt + (size_t)(obase + n)      * KC;
    const float* w1 = Wt + (size_t)(obase + 16 + n) * KC;

    // neighbor list base per N-tile (per-lane vertex)
    const int* nv0 = neigh + (size_t)(vbase + 0 + vsel) * K_NEIGH;
    const int* nv1 = neigh + (size_t)(vbase + 2 + vsel) * K_NEIGH;
    const int* nv2 = neigh + (size_t)(vbase + 4 + vsel) * K_NEIGH;

    v8f a00 = {}, a01 = {}, a02 = {}, a10 = {}, a11 = {}, a12 = {};

    const int rowStride = 8 * Cin;       // floats per vertex in (V,8,C)
    for (int k = 0; k < K_NEIGH; ++k) {
        const float* xv0 = xin + (size_t)nv0[k] * rowStride + bb * Cin;
        const float* xv1 = xin + (size_t)nv1[k] * rowStride + bb * Cin;
        const float* xv2 = xin + (size_t)nv2[k] * rowStride + bb * Cin;
        const float* wk0 = w0 + k * Cin;
        const float* wk1 = w1 + k * Cin;
        for (int c0 = 0; c0 < Cin; c0 += 4) {
            const int c = c0 + half * 2;             // this lane's 2 kc values
            v2f fa0 = *(const v2f*)(wk0 + c);
            v2f fa1 = *(const v2f*)(wk1 + c);
            v2f fb0 = *(const v2f*)(xv0 + c);
            v2f fb1 = *(const v2f*)(xv1 + c);
            v2f fb2 = *(const v2f*)(xv2 + c);
            a00 = __builtin_amdgcn_wmma_f32_16x16x4_f32(false, fa0, false, fb0, (short)0, a00, false, false);
            a01 = __builtin_amdgcn_wmma_f32_16x16x4_f32(false, fa0, false, fb1, (short)0, a01, false, false);
            a02 = __builtin_amdgcn_wmma_f32_16x16x4_f32(false, fa0, false, fb2, (short)0, a02, false, false);
            a10 = __builtin_amdgcn_wmma_f32_16x16x4_f32(false, fa1, false, fb0, (short)0, a10, false, false);
            a11 = __builtin_amdgcn_wmma_f32_16x16x4_f32(false, fa1, false, fb1, (short)0, a11, false, false);
            a12 = __builtin_amdgcn_wmma_f32_16x16x4_f32(false, fa1, false, fb2, (short)0, a12, false, false);
        }
    }

    // D layout: VGPR j: lanes 0-15 -> M=j, lanes 16-31 -> M=j+8; N = lane&15.
    // In (V,8,C) the lane's 8 outputs per tile are CONSECUTIVE channels.
    float psum[2][8], psq[2][8];
    #pragma unroll
    for (int mt = 0; mt < 2; ++mt)
        #pragma unroll
        for (int j = 0; j < 8; ++j) { psum[mt][j] = 0.f; psq[mt][j] = 0.f; }

    const int ochan0 = obase + half * 8;  // first channel of this lane's run
    #pragma unroll
    for (int mt = 0; mt < 2; ++mt) {
        const int oc = ochan0 + mt * 16;
        v4f blo, bhi;
        #pragma unroll
        for (int nt = 0; nt < 3; ++nt) {
            const v8f& acc = (mt == 0) ? (nt == 0 ? a00 : nt == 1 ? a01 : a02)
                                       : (nt == 0 ? a10 : nt == 1 ? a11 : a12);
            #pragma unroll
            for (int j = 0; j < 8; ++j) {
                float val = acc[j] + bias[oc + j];
                psum[mt][j] += val;
                psq[mt][j]  += val * val;
                if (j < 4) blo[j] = val; else bhi[j - 4] = val;
            }
            const int v = vbase + nt * 2 + vsel;
            float* op = out + (size_t)v * 8 * Cout + bb * Cout + oc;
            *(v4f*)(op)     = blo;
            *(v4f*)(op + 4) = bhi;
        }
    }

    // Reduce the 16 columns inside each half-wave (xor masks < 16 stay in
    // the half), then 2 atomics per channel from lanes n==0.
    #pragma unroll
    for (int mt = 0; mt < 2; ++mt) {
        #pragma unroll
        for (int j = 0; j < 8; ++j) {
            float s  = psum[mt][j];
            float sq = psq[mt][j];
            #pragma unroll
            for (int m = 1; m < 16; m <<= 1) {
                s  += __shfl_xor(s,  m, 32);
                sq += __shfl_xor(sq, m, 32);
            }
            if (n == 0) {
                const int o = ochan0 + mt * 16 + j;
                atomicAdd(&sums[o],        s);
                atomicAdd(&sums[Cout + o], sq);
            }
        }
    }
}

// ---------------------------------------------------------------------------
// Transpose network input (V, C=4, B=8) -> (V, B=8, C=4).
// ---------------------------------------------------------------------------
__global__ void transpose_x_kernel(const float* __restrict__ in,
                                   float* __restrict__ outp, int total)
{
    int idx = blockIdx.x * blockDim.x + threadIdx.x;
    if (idx >= total) return;
    const int c = idx & 3;
    const int b = (idx >> 2) & 7;
    const int v = idx >> 5;
    outp[idx] = in[v * 32 + c * 8 + b];
}

// ---------------------------------------------------------------------------
// In-place batch-norm (stats over V*B per channel) + leaky ReLU. (V,8,C)
// ---------------------------------------------------------------------------
__global__ void bn_lrelu_kernel(float* __restrict__ data,
                                const float* __restrict__ sums,
                                const float* __restrict__ gamma,
                                const float* __restrict__ beta,
                                int total, int Cout, int V)
{
    int idx = blockIdx.x * blockDim.x + threadIdx.x;
    if (idx >= total) return;
    const int o = idx & (Cout - 1);          // channels innermost; Cout pow2
    const float inv_cnt = 1.0f / ((float)V * 8.0f);
    const float mean = sums[o] * inv_cnt;
    const float var  = sums[Cout + o] * inv_cnt - mean * mean;
    const float g    = gamma[o] * rsqrtf(var + BN_EPS);
    float val = (data[idx] - mean) * g + beta[o];
    data[idx] = (val >= 0.0f) ? val : LRELU_SLOPE * val;
}

// ---------------------------------------------------------------------------
// Pool: out[vc,b,c] = mean_k in[neigh_fine[vc*19+k], b, c]   (coalesced in c)
// ---------------------------------------------------------------------------
__global__ void pool_mean_kernel(const float* __restrict__ in,
                                 const int*   __restrict__ neigh_fine,
                                 float*       __restrict__ outp,
                                 int total, int lc)      // lc = log2(C)
{
    int idx = blockIdx.x * blockDim.x + threadIdx.x;
    if (idx >= total) return;
    const int c  = idx & ((1 << lc) - 1);
    const int b  = (idx >> lc) & 7;
    const int vc = idx >> (lc + 3);
    const int* nv = neigh_fine + (size_t)vc * K_NEIGH;
    float s = 0.0f;
    #pragma unroll
    for (int k = 0; k < K_NEIGH; ++k)
        s += in[(((size_t)nv[k] * 8 + b) << lc) + c];
    outp[idx] = s * (1.0f / (float)K_NEIGH);
}

// ---------------------------------------------------------------------------
// Final head on (V, 8, C=256): flat[b, v*256+c] = h[v*2048 + b*256 + c]
//   out[b] = sum_i flat[b,i]*w_out[i] + y[b]*w_out[V*C] + b_out
// ---------------------------------------------------------------------------
__global__ __launch_bounds__(256)
void final_dot_kernel(const float* __restrict__ h,
                      const float* __restrict__ y,
                      const float* __restrict__ w_out,
                      const float* __restrict__ b_out,
                      float*       __restrict__ out,
                      int n)                      // n = V*256
{
    const int b   = blockIdx.x;                  // one block per batch element
    const int tid = threadIdx.x;
    float acc = 0.0f;
    for (int i = tid; i < n; i += 256) {
        const int v = i >> 8, c = i & 255;
        acc += h[(size_t)v * 2048 + b * 256 + c] * w_out[i];
    }
    __shared__ float red[256];
    red[tid] = acc;
    __syncthreads();
    for (int s = 128; s > 0; s >>= 1) {
        if (tid < s) red[tid] += red[tid + s];
        __syncthreads();
    }
    if (tid == 0)
        out[b] = red[0] + y[b] * w_out[n] + b_out[0];
}

// ---------------------------------------------------------------------------
// Host-side orchestration.
// Input order: 0:x 1:y 2..33:params(4 blk x 2 layer x {W,b,gamma,beta})
//              34:w_out 35:b_out 36..39:neigh0..3
// ---------------------------------------------------------------------------
extern "C" void kernel_launch(void* const* d_in, const int* in_sizes, int n_in,
                              void* d_out, int out_size, void* d_ws, size_t ws_size,
                              hipStream_t stream)
{
    (void)in_sizes; (void)n_in; (void)out_size; (void)ws_size;

    static const int VS[4]  = {40962, 10242, 2562, 642};
    static const int CHS[5] = {4, 32, 64, 128, 256};

    const float* x     = (const float*)d_in[0];
    const float* y     = (const float*)d_in[1];
    const float* w_out = (const float*)d_in[34];
    const float* b_out = (const float*)d_in[35];
    const int* neighs[4] = {(const int*)d_in[36], (const int*)d_in[37],
                            (const int*)d_in[38], (const int*)d_in[39]};

    const size_t maxElems = (size_t)40962 * 32 * 8;   // largest activation
    float* bufA = (float*)d_ws;
    float* bufB = bufA + maxElems;
    float* sums = bufB + maxElems;                    // 2*256 floats max

    // Transpose input (V,4,8)->(V,8,4) into bufB (overwritten later, safely
    // after its only consumer — the first conv — has run).
    {
        const int total = VS[0] * 32;
        transpose_x_kernel<<<(total + 255) / 256, 256, 0, stream>>>(x, bufB, total);
    }

    const float* cur = bufB;   // current activations (read)
    float* nxt = bufA;         // where the next op writes

    for (int i = 0; i < 4; ++i) {
        const int V = VS[i];
        if (i > 0) {
            const int C = CHS[i];
            const int total = V * C * 8;
            const int lc = __builtin_ctz(C);
            pool_mean_kernel<<<(total + 255) / 256, 256, 0, stream>>>(
                cur, neighs[i - 1], nxt, total, lc);
            cur = nxt;
            nxt = (nxt == bufA) ? bufB : bufA;
        }
        for (int l = 0; l < 2; ++l) {
            const int Cin  = (l == 0) ? CHS[i] : CHS[i + 1];
            const int Cout = CHS[i + 1];
            const int p = 2 + (i * 2 + l) * 4;
            const float* W     = (const float*)d_in[p + 0];
            const float* bias  = (const float*)d_in[p + 1];
            const float* gamma = (const float*)d_in[p + 2];
            const float* beta  = (const float*)d_in[p + 3];

            hipMemsetAsync(sums, 0, 2 * (size_t)Cout * sizeof(float), stream);

            dim3 grid(V / 6, Cout / 32);   // all V divisible by 6, Cout by 32
            conv_wmma_kernel<<<grid, 32, 0, stream>>>(
                cur, neighs[i], W, bias, nxt, sums, V, Cin, Cout);

            const int total = V * Cout * 8;
            bn_lrelu_kernel<<<(total + 255) / 256, 256, 0, stream>>>(
                nxt, sums, gamma, beta, total, Cout, V);

            cur = nxt;
            nxt = (nxt == bufA) ? bufB : bufA;
        }
    }

    final_dot_kernel<<<8, 256, 0, stream>>>(
        cur, y, w_out, b_out, (float*)d_out, VS[3] * CHS[4]);
}